// CrossCompressUnit_81793357185691
// MI455X (gfx1250) — compile-verified
//
#include <hip/hip_runtime.h>

typedef float v2f __attribute__((ext_vector_type(2)));
typedef float v8f __attribute__((ext_vector_type(8)));

#define DIM        128
#define TILE        64   // batch rows per block
#define NTHREADS   128   // 4 wave32 waves -> 4 x 16 = 64 rows via WMMA

__global__ __launch_bounds__(NTHREADS)
void cross_compress_kernel(const float* __restrict__ v,
                           const float* __restrict__ e,
                           const float* __restrict__ w_vv,
                           const float* __restrict__ w_ev,
                           const float* __restrict__ w_ve,
                           const float* __restrict__ w_ee,
                           const float* __restrict__ bias_v,
                           const float* __restrict__ bias_e,
                           float* __restrict__ v_out,
                           float* __restrict__ e_out)
{
    __shared__ float s_dvv[TILE];   // e[b] . w_vv
    __shared__ float s_dev[TILE];   // v[b] . w_ev
    __shared__ float s_dve[TILE];   // e[b] . w_ve
    __shared__ float s_dee[TILE];   // v[b] . w_ee

    const int tid  = threadIdx.x;
    const int lane = tid & 31;
    const int wave = tid >> 5;               // 0..3
    const int row0 = blockIdx.x * TILE;

    // ---------------- Phase 1: 4 dots per row via V_WMMA_F32_16X16X4_F32 ----
    // A-matrix (16x4 f32) lane layout: lane l -> row M = l%16,
    //   VGPR j -> K = (l/16)*2 + j.  B (4x16) mirrored: lane l -> col N = l%16.
    const int   m     = lane & 15;           // row-in-tile (A) / column (B, D)
    const int   khalf = (lane >> 4) << 1;    // 0 or 2: K-half of the 4-chunk
    const long  rowOff = (long)(row0 + wave * 16 + m) * DIM;

    const float isC0 = (m == 0) ? 1.0f : 0.0f;   // branch-free column masks:
    const float isC1 = (m == 1) ? 1.0f : 0.0f;   // keep EXEC all-ones for WMMA

    v8f dE = {};   // D columns: 0 -> e.w_vv, 1 -> e.w_ve
    v8f dV = {};   // D columns: 0 -> v.w_ev, 1 -> v.w_ee

    #pragma unroll 4
    for (int kc = 0; kc < DIM / 4; ++kc) {
        const int kbase = kc * 4 + khalf;

        v2f aE = *(const v2f*)(e + rowOff + kbase);   // global_load_b64
        v2f aV = *(const v2f*)(v + rowOff + kbase);

        // B fragments: col 0 / col 1 carry the weight vectors, rest zero.
        v2f bE, bV;
        bE.x = isC0 * w_vv[kbase]     + isC1 * w_ve[kbase];
        bE.y = isC0 * w_vv[kbase + 1] + isC1 * w_ve[kbase + 1];
        bV.x = isC0 * w_ev[kbase]     + isC1 * w_ee[kbase];
        bV.y = isC0 * w_ev[kbase + 1] + isC1 * w_ee[kbase + 1];

        dE = __builtin_amdgcn_wmma_f32_16x16x4_f32(false, aE, false, bE,
                                                   (short)0, dE, false, false);
        dV = __builtin_amdgcn_wmma_f32_16x16x4_f32(false, aV, false, bV,
                                                   (short)0, dV, false, false);
    }

    // D layout: lane l, VGPR r -> (M = r + 8*(l/16), N = l%16).
    // Lanes with N==0 hold column-0 dots, N==1 hold column-1 dots.
    if (m < 2) {
        const int mbase = wave * 16 + ((lane >> 4) << 3);
        #pragma unroll
        for (int r = 0; r < 8; ++r) {
            const int row = mbase + r;
            if (m == 0) { s_dvv[row] = dE[r]; s_dev[row] = dV[r]; }
            else        { s_dve[row] = dE[r]; s_dee[row] = dV[r]; }
        }
    }
    __syncthreads();

    // ---------------- Phase 2: bandwidth-bound streaming combine ------------
    // v_out[b,c] = v[b,c]*dvv[b] + e[b,c]*dev[b] + bias_v[c]
    // e_out[b,c] = v[b,c]*dve[b] + e[b,c]*dee[b] + bias_e[c]
    const float4* v4    = (const float4*)v;
    const float4* e4    = (const float4*)e;
    const float4* bv4   = (const float4*)bias_v;
    const float4* be4   = (const float4*)bias_e;
    float4*       vout4 = (float4*)v_out;
    float4*       eout4 = (float4*)e_out;

    const int C4    = DIM / 4;            // 32 float4 per row
    const int items = TILE * C4;          // 2048 per block

    for (int idx = tid; idx < items; idx += NTHREADS) {
        const int  r = idx >> 5;          // row within tile
        const int  c = idx & 31;          // float4 column
        const long g = (long)(row0 + r) * C4 + c;

        const float4 vv = v4[g];          // global_load_b128
        const float4 ee = e4[g];
        const float4 bv = bv4[c];
        const float4 be = be4[c];

        const float dvv = s_dvv[r], dev = s_dev[r];
        const float dve = s_dve[r], dee = s_dee[r];

        float4 o1, o2;
        o1.x = fmaf(vv.x, dvv, fmaf(ee.x, dev, bv.x));
        o1.y = fmaf(vv.y, dvv, fmaf(ee.y, dev, bv.y));
        o1.z = fmaf(vv.z, dvv, fmaf(ee.z, dev, bv.z));
        o1.w = fmaf(vv.w, dvv, fmaf(ee.w, dev, bv.w));
        o2.x = fmaf(vv.x, dve, fmaf(ee.x, dee, be.x));
        o2.y = fmaf(vv.y, dve, fmaf(ee.y, dee, be.y));
        o2.z = fmaf(vv.z, dve, fmaf(ee.z, dee, be.z));
        o2.w = fmaf(vv.w, dve, fmaf(ee.w, dee, be.w));

        vout4[g] = o1;                    // global_store_b128
        eout4[g] = o2;
    }
}

extern "C" void kernel_launch(void* const* d_in, const int* in_sizes, int n_in,
                              void* d_out, int out_size, void* d_ws, size_t ws_size,
                              hipStream_t stream) {
    const float* v      = (const float*)d_in[0];
    const float* e      = (const float*)d_in[1];
    const float* w_vv   = (const float*)d_in[2];
    const float* w_ev   = (const float*)d_in[3];
    const float* w_ve   = (const float*)d_in[4];
    const float* w_ee   = (const float*)d_in[5];
    const float* bias_v = (const float*)d_in[6];
    const float* bias_e = (const float*)d_in[7];

    const int B = in_sizes[0] / DIM;      // 16384
    float* v_out = (float*)d_out;
    float* e_out = v_out + (size_t)B * DIM;

    dim3 grid(B / TILE);                  // 256 blocks
    dim3 block(NTHREADS);
    cross_compress_kernel<<<grid, block, 0, stream>>>(
        v, e, w_vv, w_ev, w_ve, w_ee, bias_v, bias_e, v_out, e_out);
}